// EnCTCLoss_11965778886981
// MI455X (gfx1250) — compile-verified
//
#include <hip/hip_runtime.h>
#include <math.h>

#define T_DIM 64
#define S_DIM 20
#define EPS_NAN (-100000000.0f)
#define EPS_LSE 1e-26f
#define EPS_V 1e-06f
#define H_RATE 0.2f
#define UNI_RATE 1.5f

typedef __attribute__((ext_vector_type(2))) float v2f;
typedef __attribute__((ext_vector_type(8))) float v8f;

// masked 2-way LSE, faithful to the reference's EPS_NAN handling (branchless)
__device__ __forceinline__ float lse2(float a, float b) {
  float m = fmaxf(a, b);
  float ea = __expf(a - m);
  float eb = __expf(b - m);
  float s = ((a > EPS_NAN) ? ea : 0.0f) + ((b > EPS_NAN) ? eb : 0.0f);
  return (m > EPS_NAN) ? (__logf(s + EPS_LSE) + m) : EPS_NAN;
}

__device__ __forceinline__ float lse3(float a, float b, float c) {
  float m = fmaxf(fmaxf(a, b), c);
  float s = ((a > EPS_NAN) ? __expf(a - m) : 0.0f) +
            ((b > EPS_NAN) ? __expf(b - m) : 0.0f) +
            ((c > EPS_NAN) ? __expf(c - m) : 0.0f);
  return (m > EPS_NAN) ? (__logf(s + EPS_LSE) + m) : EPS_NAN;
}

__global__ __launch_bounds__(256) void enctc_kernel(
    const float* __restrict__ pred, const int* __restrict__ target,
    const int* __restrict__ pred_length, const int* __restrict__ target_length,
    float* __restrict__ partial, int B, int C, int S) {
  __shared__ float sh_betas[T_DIM][S_DIM][2];
  __shared__ float sh_bent[T_DIM][S_DIM][2];
  __shared__ float sh_alph[T_DIM][S_DIM];
  __shared__ float sh_aent[T_DIM][S_DIM];
  __shared__ float sh_pg1[T_DIM][S_DIM];   // pred[t, b, target[b,s]]
  __shared__ float sh_v1[T_DIM][S_DIM];    // scratch: values, then exp contribs
  __shared__ float sh_v2[T_DIM][S_DIM];
  __shared__ float sh_pbp[T_DIM];          // pred[t, b, BLANK]
  __shared__ float sh_lab[T_DIM], sh_lentv[T_DIM];
  __shared__ float sh_max1[S_DIM], sh_max2[S_DIM];
  __shared__ float sh_sum1[S_DIM], sh_sum2[S_DIM];
  __shared__ int sh_tgt[S_DIM];

  const int b = blockIdx.x;
  const int tid = threadIdx.x;
  const int pl = pred_length[b];
  const int tl = target_length[b];
  const int tl1 = tl - 1;
  const int uni_len =
      (int)floorf(UNI_RATE * (float)pl / ((float)tl + 1.0f));

  __builtin_prefetch(pred + (size_t)b * C, 0, 1);  // global_prefetch

  if (tid < S) sh_tgt[tid] = target[b * S + tid];
  if (tid >= 64 && tid < 128) {
    int t = tid - 64;
    sh_pbp[t] = pred[((size_t)t * B + b) * C];  // BLANK = 0
  }
  __syncthreads();
  for (int idx = tid; idx < T_DIM * S; idx += 256) {
    int t = idx / S, s = idx - t * S;
    sh_pg1[t][s] = pred[((size_t)t * B + b) * C + sh_tgt[s]];
  }
  __syncthreads();

  // ---- t = 0 init ----
  if (tid < S) {
    float p0 = sh_pbp[0], pc = sh_pg1[0][tid];
    sh_betas[0][tid][0] = p0;
    sh_betas[0][tid][1] = pc;
    sh_bent[0][tid][0] = p0 + __logf(-p0 + EPS_V);
    sh_bent[0][tid][1] = pc + __logf(-pc + EPS_V);
    float a0 = (tid == 0) ? sh_pg1[0][0] : EPS_NAN;
    sh_alph[0][tid] = a0;
    sh_aent[0][tid] = (tid == 0) ? (a0 + __logf(-a0 + EPS_V)) : EPS_NAN;
  }
  __syncthreads();
  if (tid == 0) {
    // umf[0] = (0 >= T - uni_len)  <=>  uni_len >= T
    float g0 = (uni_len >= T_DIM) ? 0.0f : EPS_NAN;
    sh_lab[0] = sh_alph[0][tl1] + g0;
    sh_lentv[0] = sh_aent[0][tl1] + g0;
  }
  __syncthreads();

  // ---- main recurrence over t ----
  for (int t = 1; t < T_DIM; ++t) {
    const float p0 = sh_pbp[t];
    // Phase A: update betas[:t], bent[:t]; append row t
    for (int idx = tid; idx < t * S; idx += 256) {
      int i = idx / S, s = idx - i * S;
      float pc = sh_pg1[t][s];
      float b0o = sh_betas[i][s][0], b1o = sh_betas[i][s][1];
      float nb0 = b0o + p0;
      float nb1 = lse2(nb0, b1o) + p0;
      sh_betas[i][s][0] = nb0;
      sh_betas[i][s][1] = nb1;
      float be0 = sh_bent[i][s][0], be1 = sh_bent[i][s][1];
      float t10 = be0 + p0;
      float t11 = lse2(be0, be1) + pc;
      float t20 = nb0 + __logf(-p0 + EPS_V);
      float t21 = nb1 + __logf(-pc + EPS_V);
      sh_bent[i][s][0] = lse2(t10, t20);
      sh_bent[i][s][1] = lse2(t11, t21);
    }
    if (tid < S) {
      float pc = sh_pg1[t][tid];
      sh_betas[t][tid][0] = p0;
      sh_betas[t][tid][1] = pc;
      sh_bent[t][tid][0] = p0 + __logf(-p0 + EPS_V);
      sh_bent[t][tid][1] = pc + __logf(-pc + EPS_V);
    }
    __syncthreads();

    // Phase B: reduction inputs, i in [0,t), s in [0,S-1)
    for (int idx = tid; idx < t * (S - 1); idx += 256) {
      int i = idx / (S - 1), s = idx - i * (S - 1);
      float bet = sh_betas[i + 1][s + 1][1];
      float al = sh_alph[i][s];
      sh_v1[i][s] = al + bet;
      sh_v2[i][s] = lse2(sh_aent[i][s] + bet, al + sh_bent[i + 1][s + 1][1]);
    }
    __syncthreads();

    // Phase C: unmasked per-column maxima over i in [0,t)
    if (tid < 2 * (S - 1)) {
      int which = tid / (S - 1);
      int s = tid - which * (S - 1);
      const float(*v)[S_DIM] = which ? sh_v2 : sh_v1;
      float m = -INFINITY;
      for (int i = 0; i < t; ++i) m = fmaxf(m, v[i][s]);
      if (which) sh_max2[s] = m; else sh_max1[s] = m;
    }
    __syncthreads();

    // Phase C2: in-place masked exp transform (branchless), performed by all
    // 256 threads. Rows [t, t_pad) and column S-1 are force-zeroed so the
    // WMMA loop below needs no guards.
    {
      int t_pad = (t + 3) & ~3;
      int tmin = t - uni_len;
      for (int idx = tid; idx < t_pad * S; idx += 256) {
        int i = idx / S, s = idx - i * S;
        int sc = (s < S - 1) ? s : (S - 2);
        float m1 = sh_max1[sc], m2 = sh_max2[sc];
        float x1 = sh_v1[i][s], x2 = sh_v2[i][s];
        bool ok = (i < t) && (i >= tmin) && (s < S - 1);
        float e1 = __expf(x1 - m1);
        float e2 = __expf(x2 - m2);
        sh_v1[i][s] = (ok && (x1 > EPS_NAN)) ? e1 : 0.0f;
        sh_v2[i][s] = (ok && (x2 > EPS_NAN)) ? e2 : 0.0f;
      }
    }
    __syncthreads();

    // Phase D: column sums via V_WMMA_F32_16X16X4_F32, fully branchless.
    // A = all-ones (16x4) -> every row of D equals the column sums, which is
    // robust to the B-operand K-ordering. Hi-group columns >= S-1 clamp onto
    // the zeroed column S-1.
    {
      int wid = tid >> 5;
      if (wid < 2) {
        const float(*xv)[S_DIM] = wid ? sh_v2 : sh_v1;
        float* sums = wid ? sh_sum2 : sh_sum1;
        int lane = tid & 31;
        int colL = lane & 15;                        // cols 0..15
        int chi = 16 + colL;                         // cols 16..31 (virtual)
        int chi_c = (chi < S - 1) ? chi : (S - 1);   // col S-1 is all zeros
        int kofs = (lane >> 4) << 1;                 // rows k0,k0+1 per half
        v2f ones;
        ones[0] = 1.0f;
        ones[1] = 1.0f;
        v8f acc_lo = {};
        v8f acc_hi = {};
        for (int kb = 0; kb < t; kb += 4) {
          int k0 = kb + kofs;
          v2f blo, bhi;
          blo[0] = xv[k0][colL];
          blo[1] = xv[k0 + 1][colL];
          bhi[0] = xv[k0][chi_c];
          bhi[1] = xv[k0 + 1][chi_c];
          acc_lo = __builtin_amdgcn_wmma_f32_16x16x4_f32(
              false, ones, false, blo, (short)0, acc_lo, false, false);
          acc_hi = __builtin_amdgcn_wmma_f32_16x16x4_f32(
              false, ones, false, bhi, (short)0, acc_hi, false, false);
        }
        if (lane < 16) sums[colL] = acc_lo[0];
        else if (chi < S - 1) sums[chi] = acc_hi[0];
      }
    }
    __syncthreads();

    // Phase E: alphas[t], aent[t]
    if (tid < S) {
      float a, ae;
      if (tid == 0) {
        // umf[T-t] = (T-t >= T - uni_len) <=> uni_len >= t
        float gate = (uni_len >= t) ? 0.0f : EPS_NAN;
        a = sh_betas[0][0][1] + gate;
        ae = sh_bent[0][0][1] + gate;
      } else {
        int c = tid - 1;
        bool anym = (uni_len >= 1);  // exists i in [0,t): i >= t-uni_len
        bool n1 = (sh_max1[c] > EPS_NAN) && anym;
        bool n2 = (sh_max2[c] > EPS_NAN) && anym;
        a = n1 ? (__logf(sh_sum1[c] + EPS_LSE) + sh_max1[c]) : EPS_NAN;
        ae = n2 ? (__logf(sh_sum2[c] + EPS_LSE) + sh_max2[c]) : EPS_NAN;
      }
      sh_alph[t][tid] = a;
      sh_aent[t][tid] = ae;
    }
    __syncthreads();

    // Phase F: labels / lent
    if (tid == 0) {
      // umf[t] = (t >= T - uni_len) <=> uni_len >= T - t
      float gate = (uni_len >= T_DIM - t) ? 0.0f : EPS_NAN;
      float pb = sh_pbp[t];
      float x1 = sh_lab[t - 1] + pb + gate;
      float newlab = lse2(x1, sh_alph[t][tl1]);
      float y1 = sh_lentv[t - 1] + pb + gate;
      float y2 = sh_lab[t - 1] + pb + __logf(-pb + EPS_V) + gate;
      sh_lentv[t] = lse3(y1, y2, sh_aent[t][tl1]);
      sh_lab[t] = newlab;
    }
    __syncthreads();
  }

  if (tid == 0) {
    float lt = sh_lab[pl - 1];
    float lte = sh_lentv[pl - 1];
    float H = __expf(lte - lt) + lt;
    partial[b] = (1.0f - H_RATE) * (-lt) - H_RATE * H;
  }
}

__global__ void enctc_reduce_kernel(const float* __restrict__ partial,
                                    float* __restrict__ out, int B) {
  if (threadIdx.x == 0 && blockIdx.x == 0) {
    float s = 0.0f;
    for (int i = 0; i < B; ++i) s += partial[i];  // fixed order: deterministic
    out[0] = s;
  }
}

extern "C" void kernel_launch(void* const* d_in, const int* in_sizes, int n_in,
                              void* d_out, int out_size, void* d_ws,
                              size_t ws_size, hipStream_t stream) {
  const float* pred = (const float*)d_in[0];
  const int* target = (const int*)d_in[1];
  const int* pred_length = (const int*)d_in[2];
  const int* target_length = (const int*)d_in[3];
  const int B = in_sizes[2];                 // 96
  const int S = in_sizes[1] / B;             // 20
  const int C = in_sizes[0] / (T_DIM * B);   // 128 (T fixed at 64)
  float* partial = (float*)d_ws;             // B floats of scratch

  enctc_kernel<<<B, 256, 0, stream>>>(pred, target, pred_length, target_length,
                                      partial, B, C, S);
  enctc_reduce_kernel<<<1, 32, 0, stream>>>(partial, (float*)d_out, B);
}